// ROUGELoss_48052094107966
// MI455X (gfx1250) — compile-verified
//
#include <hip/hip_runtime.h>
#include <math.h>

typedef float v2f __attribute__((ext_vector_type(2)));
typedef float v8f __attribute__((ext_vector_type(8)));

#define T_LOG 512
#define T_LAB 512
#define VOCAB 32000
#define BATCH 4

__device__ __forceinline__ float waveReduceSum(float v) {
    for (int m = 16; m >= 1; m >>= 1) v += __shfl_xor(v, m, 32);
    return v;
}
__device__ __forceinline__ float waveReduceMax(float v) {
    for (int m = 16; m >= 1; m >>= 1) v = fmaxf(v, __shfl_xor(v, m, 32));
    return v;
}

// Phase 1: one block per (b, j) row of logits. Stream 128KB row once:
// denom = sum(exp(x)) over vocab; then gather exp(x[label[i]])/denom for the
// 512 labels (row is hot in L2), store transposed overlap T[b][j][i] and
// rowmax over i (== reference col_max at column j).
__global__ __launch_bounds__(256) void rouge_phase1(
    const float* __restrict__ logits, const int* __restrict__ labels,
    float* __restrict__ T, float* __restrict__ rowmax)
{
    const int row = blockIdx.x;            // b*512 + j
    const int b   = row >> 9;
    const int tid = threadIdx.x;
    const float*  rp  = logits + (size_t)row * VOCAB;
    const float4* rp4 = (const float4*)rp;

    __shared__ float red[8];
    __shared__ float sTotal;
    __shared__ int   lab[T_LAB];

    lab[tid]       = labels[b * T_LAB + tid];
    lab[tid + 256] = labels[b * T_LAB + tid + 256];

    float s = 0.f;
    for (int p = tid; p < VOCAB / 4; p += 256) {
        float4 x = rp4[p];
        s += __expf(x.x) + __expf(x.y) + __expf(x.z) + __expf(x.w);
    }
    s = waveReduceSum(s);
    if ((tid & 31) == 0) red[tid >> 5] = s;
    __syncthreads();
    if (tid == 0) {
        float t = 0.f;
        for (int w = 0; w < 8; ++w) t += red[w];
        sTotal = t;
    }
    __syncthreads();
    const float denom = sTotal;

    float m = -1.f;
    float* Trow = T + (size_t)row * T_LAB;
    for (int i = tid; i < T_LAB; i += 256) {
        float ov = __expf(rp[lab[i]]) / denom;
        Trow[i] = ov;
        m = fmaxf(m, ov);
    }
    m = waveReduceMax(m);
    if ((tid & 31) == 0) red[tid >> 5] = m;
    __syncthreads();
    if (tid == 0) {
        float t = red[0];
        for (int w = 1; w < 8; ++w) t = fmaxf(t, red[w]);
        rowmax[row] = t;
    }
}

// Phase 2: colmax[b][i] = max over j of T[b][j][i]  (== reference row_max).
__global__ __launch_bounds__(256) void rouge_colmax(
    const float* __restrict__ T, float* __restrict__ colmax)
{
    const int b = blockIdx.x >> 1;
    const int i = ((blockIdx.x & 1) << 8) + threadIdx.x;
    const float* Tb = T + (size_t)b * T_LOG * T_LAB;
    float m = -1.f;
    for (int j = 0; j < T_LOG; ++j) m = fmaxf(m, Tb[j * T_LAB + i]);
    colmax[b * T_LAB + i] = m;
}

// Phase 3: masked sum via V_WMMA_F32_16X16X4_F32 (D = ones(16x4) x B(4x16) + C).
// Summation is order-independent, so each lane deposits 2 arbitrary masked f32
// values per iteration into the B operand; A=ones makes every D row the
// per-column partial sum. Mask is computed BRANCHLESSLY (non-short-circuit &)
// so EXEC stays all-ones through every v_wmma, per the ISA requirement.
__global__ __launch_bounds__(256) void rouge_masked_sum(
    const float* __restrict__ T, const float* __restrict__ rowmax,
    const float* __restrict__ colmax, float* __restrict__ partials)
{
    const int b     = blockIdx.x >> 3;
    const int chunk = blockIdx.x & 7;
    const int tid   = threadIdx.x;
    const int wave  = tid >> 5, lane = tid & 31;

    __shared__ float rm[T_LOG];
    __shared__ float cm[T_LAB];
    __shared__ float wsum[8];

    rm[tid]       = rowmax[b * T_LOG + tid];
    rm[tid + 256] = rowmax[b * T_LOG + tid + 256];
    cm[tid]       = colmax[b * T_LAB + tid];
    cm[tid + 256] = colmax[b * T_LAB + tid + 256];
    __syncthreads();

    const float* Tb = T + (size_t)b * (T_LOG * T_LAB);
    const int base = chunk * 32768 + wave * 4096;

    const v2f onesA = {1.0f, 1.0f};
    v8f acc = {};
    for (int t = 0; t < 64; ++t) {
        const int e = base + t * 64 + lane * 2;       // 64 elems/iter, one row j
        const int j = e >> 9;
        const int i = e & 511;
        float2 x = *(const float2*)(Tb + e);
        const float rmj  = rm[j];
        const float cmi0 = cm[i];
        const float cmi1 = cm[i + 1];
        // branchless: evaluate all compares, combine with bitwise &
        const int k0 = (int)(x.x == rmj) & (int)(x.x == cmi0);
        const int k1 = (int)(x.y == rmj) & (int)(x.y == cmi1);
        v2f bv;
        bv.x = k0 ? x.x : 0.0f;
        bv.y = k1 ? x.y : 0.0f;
        acc = __builtin_amdgcn_wmma_f32_16x16x4_f32(
            /*neg_a=*/false, onesA, /*neg_b=*/false, bv,
            /*c_mod=*/(short)0, acc, /*reuse_a=*/false, /*reuse_b=*/false);
    }
    // All D rows identical; lanes 16..31 duplicate cols of lanes 0..15 -> x0.5.
    float part = waveReduceSum(acc[0]) * 0.5f;
    if (lane == 0) wsum[wave] = part;
    __syncthreads();
    if (tid == 0) {
        float t = 0.f;
        for (int w = 0; w < 8; ++w) t += wsum[w];
        partials[blockIdx.x] = t;
    }
}

__global__ void rouge_finalize(const float* __restrict__ partials,
                               float* __restrict__ out)
{
    const int b = threadIdx.x;
    if (b < BATCH) {
        float s = 0.f;
        for (int c = 0; c < 8; ++c) s += partials[b * 8 + c];
        // 2 * numerator / ((T_LAB - N + 1) + (T_LOG - N + 1)) = numer / 512
        out[b] = s * (1.0f / 512.0f);
    }
}

extern "C" void kernel_launch(void* const* d_in, const int* in_sizes, int n_in,
                              void* d_out, int out_size, void* d_ws, size_t ws_size,
                              hipStream_t stream)
{
    const float* logits = (const float*)d_in[0];
    const int*   labels = (const int*)d_in[1];

    float* T        = (float*)d_ws;                         // 4*512*512 f32 = 4MB
    float* rowmax   = T + (size_t)BATCH * T_LOG * T_LAB;    // 2048 f32
    float* colmax   = rowmax + BATCH * T_LOG;               // 2048 f32
    float* partials = colmax + BATCH * T_LAB;               // 32 f32

    rouge_phase1    <<<BATCH * T_LOG, 256, 0, stream>>>(logits, labels, T, rowmax);
    rouge_colmax    <<<BATCH * 2,     256, 0, stream>>>(T, colmax);
    rouge_masked_sum<<<BATCH * 8,     256, 0, stream>>>(T, rowmax, colmax, partials);
    rouge_finalize  <<<1, 32, 0, stream>>>(partials, (float*)d_out);
}